// GraphSAGELayer_80934363726337
// MI455X (gfx1250) — compile-verified
//
#include <hip/hip_runtime.h>

typedef __attribute__((ext_vector_type(2))) float v2f;
typedef __attribute__((ext_vector_type(8))) float v8f;

#define FEAT 128
#define KSTEPS 32                 // 128 / 4
#define FRAGS_PER_MAT (KSTEPS * 8 * 32)  // [kstep][ntile][lane] v2f = 8192

// ---------------------------------------------------------------------------
// Phase 0: zero agg (aliased to d_out) and deg (in d_ws)
// ---------------------------------------------------------------------------
__global__ void sage_zero_kernel(float* __restrict__ agg, float* __restrict__ deg,
                                 int n_nodes) {
  long long idx = (long long)blockIdx.x * blockDim.x + threadIdx.x;
  long long total = (long long)n_nodes * FEAT;
  if (idx < total) agg[idx] = 0.0f;
  if (idx < n_nodes) deg[idx] = 0.0f;
}

// ---------------------------------------------------------------------------
// Phase 1: scatter-add  agg[dst] += x[src], deg[dst] += 1
// One wave32 per edge: lane l handles features [4l, 4l+4).
// ---------------------------------------------------------------------------
__launch_bounds__(256)
__global__ void sage_scatter_kernel(const float* __restrict__ x,
                                    const int* __restrict__ ei,
                                    float* __restrict__ agg,
                                    float* __restrict__ deg,
                                    int n_edges) {
  int gtid = blockIdx.x * blockDim.x + threadIdx.x;
  int edge = gtid >> 5;
  int lane = gtid & 31;
  if (edge >= n_edges) return;

  int src = ei[edge];            // edge_index[0][e]
  int dst = ei[n_edges + edge];  // edge_index[1][e]

  const float4* xrow = reinterpret_cast<const float4*>(x + (long long)src * FEAT);
  float4 v = xrow[lane];

  float* arow = agg + (long long)dst * FEAT + lane * 4;
  unsafeAtomicAdd(arow + 0, v.x);
  unsafeAtomicAdd(arow + 1, v.y);
  unsafeAtomicAdd(arow + 2, v.z);
  unsafeAtomicAdd(arow + 3, v.w);
  if (lane == 0) unsafeAtomicAdd(deg + dst, 1.0f);
}

// ---------------------------------------------------------------------------
// Phase 2: fused dual GEMM + bias + ReLU with V_WMMA_F32_16X16X4_F32.
//   out[i,:] = relu(x[i,:]@Ws^T + (agg[i,:]/max(deg[i],1))@Wn^T + bs + bn)
//
// Work mapping: each wave computes a 16-row x 64-col stripe (4 n-tiles);
// waves pair up per row-tile (nhalf = wave&1 selects columns 0-63 / 64-127),
// so a 256-thread block covers 64 rows. Register footprint stays well under
// 256 VGPRs (no s_set_vgpr_msb traffic).
//
// The k-loop is FULLY UNROLLED: with no loop back-edge there is no
// accumulator phi, so the allocator keeps each accumulator in one static
// home and WMMAs accumulate strictly in place (this removes the ~64
// v_dual_mov_b32 per-iteration register-rotation block seen with a rolled
// loop). The scheduler is free to hoist ds/global loads to its preferred
// prefetch depth across the straight-line code.
//
// Both weight matrices are staged once per block into LDS in *B-fragment
// order*: frag[kstep][ntile][lane] (v2f) -> conflict-free ds_load_b64 shared
// by all waves. The two GEMMs keep SEPARATE accumulator sets (private
// def-use chains; summed with the biases in the epilogue).
//
// f32 16x16x4 fragment layout (wave32):
//   A (16x4): lane L -> M=L%16, {v0,v1} = A[M, k0 + 2*(L/16) + {0,1}]
//   B (4x16): lane L -> N=L%16, {v0,v1} = B[k0 + 2*(L/16) + {0,1}, N]
//   C:        lane L -> N=L%16, reg r = C[r + 8*(L/16), N]
// B = W^T, so B frag (s, t, L) = W[t*16 + L%16][4s + 2*(L/16) + {0,1}].
// ---------------------------------------------------------------------------
__launch_bounds__(256)
__global__ void sage_gemm_kernel(const float* __restrict__ x,
                                 const float* agg,   // aliases out (row-local RAW only)
                                 const float* __restrict__ Ws,
                                 const float* __restrict__ bs,
                                 const float* __restrict__ Wn,
                                 const float* __restrict__ bn,
                                 const float* __restrict__ deg,
                                 float* out,
                                 int n_nodes) {
  extern __shared__ char smem_raw[];
  v2f* bsf = reinterpret_cast<v2f*>(smem_raw);   // Ws fragments
  v2f* bnf = bsf + FRAGS_PER_MAT;                // Wn fragments

  const int tid = threadIdx.x;
  const int lane = tid & 31;
  const int wave = tid >> 5;
  const int nn = lane & 15;
  const int hh = lane >> 4;

  // --- Stage both weight matrices into fragment-order LDS (all waves) ---
  for (int id = tid; id < KSTEPS * FEAT; id += 256) {
    const int s = id >> 7;        // k-quad 0..31
    const int row = id & 127;     // weight row == output column
    const int t = row >> 4;
    const int nr = row & 15;
    const int base = (s * 8 + t) * 32;
    float4 w4 = *reinterpret_cast<const float4*>(Ws + row * FEAT + s * 4);
    v2f lo = {w4.x, w4.y};
    v2f hi = {w4.z, w4.w};
    bsf[base + nr] = lo;
    bsf[base + 16 + nr] = hi;
    float4 n4 = *reinterpret_cast<const float4*>(Wn + row * FEAT + s * 4);
    v2f nlo = {n4.x, n4.y};
    v2f nhi = {n4.z, n4.w};
    bnf[base + nr] = nlo;
    bnf[base + 16 + nr] = nhi;
  }
  __syncthreads();

  const int rowTile = wave >> 1;   // 0..3 within block
  const int nhalf = wave & 1;      // column half: 0 -> cols 0-63, 1 -> 64-127
  const int m0 = (blockIdx.x * 4 + rowTile) * 16;
  if (m0 >= n_nodes) return;  // wave-uniform (after barrier; EXEC all-ones below)

  // This lane's fixed A-row (clamped for tail tile; stores are guarded).
  int arow = m0 + nn;
  if (arow >= n_nodes) arow = n_nodes - 1;
  const float inv = 1.0f / fmaxf(deg[arow], 1.0f);

  const float* __restrict__ xrow = x + (long long)arow * FEAT;
  const float* grow = agg + (long long)arow * FEAT;

  v8f acc_s[4] = {};  // x @ Ws^T
  v8f acc_n[4] = {};  // (agg/deg) @ Wn^T

#pragma unroll
  for (int s = 0; s < KSTEPS; ++s) {
    const int ka = 4 * s + 2 * hh;
    v2f a_x = *reinterpret_cast<const v2f*>(xrow + ka);
    v2f a_g = *reinterpret_cast<const v2f*>(grow + ka);
    a_g *= inv;  // fold mean-normalization into A fragment

    // This wave's 8 B fragments (4 per matrix) from fragment-order LDS.
    const v2f* bsrow = bsf + (s * 8 + nhalf * 4) * 32 + lane;
    const v2f* bnrow = bnf + (s * 8 + nhalf * 4) * 32 + lane;
    v2f bsv[4], bnv[4];
#pragma unroll
    for (int t = 0; t < 4; ++t) bsv[t] = bsrow[t * 32];
#pragma unroll
    for (int t = 0; t < 4; ++t) bnv[t] = bnrow[t * 32];

    // Private accumulation chains (in-place D==C).
#pragma unroll
    for (int t = 0; t < 4; ++t)
      acc_s[t] = __builtin_amdgcn_wmma_f32_16x16x4_f32(
          false, a_x, false, bsv[t], (short)0, acc_s[t], false, false);
#pragma unroll
    for (int t = 0; t < 4; ++t)
      acc_n[t] = __builtin_amdgcn_wmma_f32_16x16x4_f32(
          false, a_g, false, bnv[t], (short)0, acc_n[t], false, false);
  }

  // Epilogue: combine both GEMMs + bias + ReLU + guarded store.
#pragma unroll
  for (int t = 0; t < 4; ++t) {
    const int ncol = nhalf * 64 + t * 16 + nn;
    const float bias = bs[ncol] + bn[ncol];
#pragma unroll
    for (int r = 0; r < 8; ++r) {
      const int row = m0 + r + 8 * hh;
      if (row < n_nodes) {
        float v = acc_s[t][r] + acc_n[t][r] + bias;
        out[(long long)row * FEAT + ncol] = fmaxf(v, 0.0f);
      }
    }
  }
}

// ---------------------------------------------------------------------------
// Host launcher (graph-capture safe: stream-ordered kernels only).
// ---------------------------------------------------------------------------
extern "C" void kernel_launch(void* const* d_in, const int* in_sizes, int n_in,
                              void* d_out, int out_size, void* d_ws, size_t ws_size,
                              hipStream_t stream) {
  const float* x  = (const float*)d_in[0];
  const int*   ei = (const int*)d_in[1];
  const float* Ws = (const float*)d_in[2];
  const float* bs = (const float*)d_in[3];
  const float* Wn = (const float*)d_in[4];
  const float* bn = (const float*)d_in[5];
  float* out = (float*)d_out;

  const int n_nodes = in_sizes[0] / FEAT;  // 100000
  const int n_edges = in_sizes[1] / 2;     // 600000

  float* agg = out;             // accumulate mean-agg numerator in-place in d_out
  float* deg = (float*)d_ws;    // n_nodes floats of scratch

  // Phase 0: zero agg + deg
  {
    long long total = (long long)n_nodes * FEAT;
    int blocks = (int)((total + 255) / 256);
    sage_zero_kernel<<<blocks, 256, 0, stream>>>(agg, deg, n_nodes);
  }
  // Phase 1: scatter-add (one wave per edge, 8 waves per 256-thread block)
  {
    int blocks = (n_edges + 7) / 8;
    sage_scatter_kernel<<<blocks, 256, 0, stream>>>(x, ei, agg, deg, n_edges);
  }
  // Phase 2: fused dual-WMMA GEMM + bias + ReLU (128 KB dynamic LDS for both
  // weight matrices in fragment order; gfx1250 WGP has 320 KB LDS).
  // Each block: 8 waves = 4 row-tiles x 2 column halves = 64 rows.
  {
    int blocks = (n_nodes + 63) / 64;
    size_t lds_bytes = 2 * FRAGS_PER_MAT * sizeof(v2f);  // 131072
    sage_gemm_kernel<<<blocks, 256, lds_bytes, stream>>>(x, agg, Ws, bs, Wn, bn,
                                                         deg, out, n_nodes);
  }
}